// SymmetricTensorProduct_52879637348694
// MI455X (gfx1250) — compile-verified
//
#include <hip/hip_runtime.h>

typedef __attribute__((ext_vector_type(2))) float        v2f;
typedef __attribute__((ext_vector_type(8))) float        v8f;
typedef __attribute__((ext_vector_type(4))) unsigned int v4u;
typedef __attribute__((ext_vector_type(8))) int          v8i;
typedef __attribute__((ext_vector_type(4))) int          v4i;

#define NUM_SEG 128
#define SEG_DIM 32
#define X0SZ    (NUM_SEG * SEG_DIM)   // 4096 floats per row
#define NP2     2048
#define NP3     4096

#if defined(__AMDGCN__) && __has_builtin(__builtin_amdgcn_tensor_load_to_lds) && \
    __has_builtin(__builtin_amdgcn_s_wait_tensorcnt)
#define STP_TDM 1
#else
#define STP_TDM 0
#endif

// ---------------------------------------------------------------------------
// Build dense degree-1 segment matrix W1[o][i] = sum of c1 over paths (i->o).
// ---------------------------------------------------------------------------
__global__ void stp_zero_w1(float* __restrict__ w1) {
    int i = blockIdx.x * blockDim.x + threadIdx.x;
    if (i < NUM_SEG * NUM_SEG) w1[i] = 0.0f;
}

__global__ void stp_build_w1(const float* __restrict__ c1,
                             const int* __restrict__ i1,
                             const int* __restrict__ o1,
                             float* __restrict__ w1, int p1) {
    int p = blockIdx.x * blockDim.x + threadIdx.x;
    if (p < p1) atomicAdd(&w1[o1[p] * NUM_SEG + i1[p]], c1[p]);
}

// ---------------------------------------------------------------------------
// Main kernel: one block per batch row, 256 threads = 8 waves (wave32).
// lane == intra-segment component k (SEG_DIM == 32 == wave width).
// ---------------------------------------------------------------------------
__global__ __launch_bounds__(256)
void stp_kernel(const float* __restrict__ x0,
                const float* __restrict__ coeff0,
                const float* __restrict__ w1,
                const float* __restrict__ c2, const int* __restrict__ a2,
                const int* __restrict__ b2, const int* __restrict__ o2,
                const float* __restrict__ c3, const int* __restrict__ a3,
                const int* __restrict__ b3, const int* __restrict__ d3i,
                const int* __restrict__ o3,
                float* __restrict__ out)
{
    __shared__ float xs[X0SZ];    // staged x row   (16 KB)
    __shared__ float acc[X0SZ];   // accumulator    (16 KB)

    const int b    = blockIdx.x;
    const int tid  = threadIdx.x;
    const int lane = tid & 31;
    const int wave = tid >> 5;

    const float* __restrict__ xrow = x0 + (size_t)b * X0SZ;
    float4* acc4 = (float4*)acc;

    // ---- phase 1: stage x[b] into LDS via TDM, zero accumulator -----------
#if STP_TDM
    if (wave == 0) {
        // Tensor DMA descriptor: 1 row x 4096 elems of 4B, global -> LDS.
        unsigned lds_off = (unsigned)(__SIZE_TYPE__)
                           (__attribute__((address_space(3))) float*)xs;
        unsigned long long ga = (unsigned long long)(__SIZE_TYPE__)xrow;
        v4u g0 = { 1u,                                    // count=1 (valid user D#)
                   lds_off,                               // lds_addr (bytes)
                   (unsigned)(ga & 0xffffffffu),          // global_addr[31:0]
                   ((unsigned)(ga >> 32) & 0x01ffffffu)   // global_addr[56:32]
                     | (2u << 30) };                      // type=2 ("image")
        v8i g1 = { (int)(2u << 16),        // wg_mask=0 | data_size=2 (4 bytes)
                   (int)(0x1000u << 16),   // tensor_dim0 = 4096 (bits 79:48, lo16)
                   (int)(1u << 16),        // tensor_dim0 hi=0 | tensor_dim1 = 1
                   (int)(0x1000u << 16),   // tensor_dim1 hi=0 | tile_dim0 = 4096
                   1,                      // tile_dim1 = 1, tile_dim2 = 0
                   0x1000,                 // tensor_dim0_stride = 4096 (lo32)
                   (int)(0x1000u << 16),   // dim0_stride hi=0 | dim1_stride lo16
                   0 };
        v4i gz  = { 0, 0, 0, 0 };          // groups 2/3 unused (2D tensor)
        v8i gz8 = { 0, 0, 0, 0, 0, 0, 0, 0 };
        __builtin_amdgcn_tensor_load_to_lds(g0, g1, gz, gz, gz8, 0);
    }
    #pragma unroll
    for (int i = tid; i < X0SZ / 4; i += 256)
        acc4[i] = make_float4(0.f, 0.f, 0.f, 0.f);   // overlap with TDM
    if (wave == 0) __builtin_amdgcn_s_wait_tensorcnt(0);
    __syncthreads();
#else
    {
        const float4* __restrict__ xrow4 = (const float4*)xrow;
        float4* xs4 = (float4*)xs;
        #pragma unroll
        for (int i = tid; i < X0SZ / 4; i += 256) {
            xs4[i]  = xrow4[i];
            acc4[i] = make_float4(0.f, 0.f, 0.f, 0.f);
        }
    }
    __syncthreads();
#endif

    // ---- phase 2: degree-1 term  acc = W1(128x128) @ xseg(128x32) ---------
    // fp32 WMMA 16x16x4. Wave w owns row-tile mt=w (A shared across both
    // column tiles); A preloaded into registers so loads clause together.
    {
        const int mt   = wave;
        const int arow = mt * 16 + (lane & 15);   // A rows 0..15 in both lane halves
        const int kb   = (lane >> 4) * 2;         // lane half selects K pair {0,1}/{2,3}
        v2f areg[32];
        #pragma unroll
        for (int kk = 0; kk < 32; ++kk)
            areg[kk] = *(const v2f*)(w1 + arow * NUM_SEG + kk * 4 + kb);

        #pragma unroll
        for (int nt = 0; nt < 2; ++nt) {
            const int ncol = nt * 16 + (lane & 15);
            v8f d = {};
            #pragma unroll
            for (int kk = 0; kk < 32; ++kk) {
                const int k0 = kk * 4 + kb;
                v2f bv = { xs[k0 * SEG_DIM + ncol], xs[(k0 + 1) * SEG_DIM + ncol] };
                d = __builtin_amdgcn_wmma_f32_16x16x4_f32(
                        /*neg_a=*/false, areg[kk], /*neg_b=*/false, bv,
                        /*c_mod=*/(short)0, d, /*reuse_a=*/false, /*reuse_b=*/false);
            }
            // D layout: VGPR r -> row (mt*16 + r + 8*(lane>=16)), col = ncol.
            const int rbase = mt * 16 + (lane >> 4) * 8;
            #pragma unroll
            for (int r = 0; r < 8; ++r)
                acc[(rbase + r) * SEG_DIM + ncol] = d[r];   // exclusive ownership
        }
    }
    __syncthreads();

    // ---- phase 3: degree-2 / degree-3 paths, 4 paths per wave iteration ---
    // Table fetches are int4/float4 (one b128 per table per 4 paths); the 12
    // LDS gathers + 4 ds_add_f32 per iteration give the scheduler ILP.
    for (int p = wave * 4; p < NP2; p += 32) {
        const int4   ai = *(const int4*)  (a2 + p);
        const int4   bi = *(const int4*)  (b2 + p);
        const int4   oi = *(const int4*)  (o2 + p);
        const float4 ci = *(const float4*)(c2 + p);
        atomicAdd(&acc[oi.x * SEG_DIM + lane],
                  ci.x * xs[ai.x * SEG_DIM + lane] * xs[bi.x * SEG_DIM + lane]);
        atomicAdd(&acc[oi.y * SEG_DIM + lane],
                  ci.y * xs[ai.y * SEG_DIM + lane] * xs[bi.y * SEG_DIM + lane]);
        atomicAdd(&acc[oi.z * SEG_DIM + lane],
                  ci.z * xs[ai.z * SEG_DIM + lane] * xs[bi.z * SEG_DIM + lane]);
        atomicAdd(&acc[oi.w * SEG_DIM + lane],
                  ci.w * xs[ai.w * SEG_DIM + lane] * xs[bi.w * SEG_DIM + lane]);
    }
    for (int p = wave * 4; p < NP3; p += 32) {
        const int4   ai = *(const int4*)  (a3 + p);
        const int4   bi = *(const int4*)  (b3 + p);
        const int4   di = *(const int4*)  (d3i + p);
        const int4   oi = *(const int4*)  (o3 + p);
        const float4 ci = *(const float4*)(c3 + p);
        atomicAdd(&acc[oi.x * SEG_DIM + lane],
                  ci.x * xs[ai.x * SEG_DIM + lane] * xs[bi.x * SEG_DIM + lane]
                       * xs[di.x * SEG_DIM + lane]);
        atomicAdd(&acc[oi.y * SEG_DIM + lane],
                  ci.y * xs[ai.y * SEG_DIM + lane] * xs[bi.y * SEG_DIM + lane]
                       * xs[di.y * SEG_DIM + lane]);
        atomicAdd(&acc[oi.z * SEG_DIM + lane],
                  ci.z * xs[ai.z * SEG_DIM + lane] * xs[bi.z * SEG_DIM + lane]
                       * xs[di.z * SEG_DIM + lane]);
        atomicAdd(&acc[oi.w * SEG_DIM + lane],
                  ci.w * xs[ai.w * SEG_DIM + lane] * xs[bi.w * SEG_DIM + lane]
                       * xs[di.w * SEG_DIM + lane]);
    }
    __syncthreads();

    // ---- phase 4: out = acc + coeff0 --------------------------------------
    float* __restrict__ orow = out + (size_t)b * X0SZ;
    #pragma unroll
    for (int i = tid; i < X0SZ; i += 256)
        orow[i] = acc[i] + coeff0[i];
}

// ---------------------------------------------------------------------------
extern "C" void kernel_launch(void* const* d_in, const int* in_sizes, int n_in,
                              void* d_out, int out_size, void* d_ws, size_t ws_size,
                              hipStream_t stream)
{
    const float* x0     = (const float*)d_in[0];
    const float* coeff0 = (const float*)d_in[1];
    const float* c1     = (const float*)d_in[2];
    const int*   i1     = (const int*)  d_in[3];
    const int*   o1     = (const int*)  d_in[4];
    const float* c2     = (const float*)d_in[5];
    const int*   a2     = (const int*)  d_in[6];
    const int*   b2     = (const int*)  d_in[7];
    const int*   o2     = (const int*)  d_in[8];
    const float* c3     = (const float*)d_in[9];
    const int*   a3     = (const int*)  d_in[10];
    const int*   b3     = (const int*)  d_in[11];
    const int*   d3i    = (const int*)  d_in[12];
    const int*   o3     = (const int*)  d_in[13];
    float*       out    = (float*)d_out;
    float*       w1     = (float*)d_ws;               // 128*128 floats = 64 KB

    const int batch = in_sizes[0] / X0SZ;
    const int p1    = in_sizes[2];

    stp_zero_w1<<<(NUM_SEG * NUM_SEG + 255) / 256, 256, 0, stream>>>(w1);
    stp_build_w1<<<(p1 + 127) / 128, 128, 0, stream>>>(c1, i1, o1, w1, p1);
    stp_kernel<<<batch, 256, 0, stream>>>(x0, coeff0, w1,
                                          c2, a2, b2, o2,
                                          c3, a3, b3, d3i, o3,
                                          out);
}